// TranslationLoss_32298154065999
// MI455X (gfx1250) — compile-verified
//
#include <hip/hip_runtime.h>
#include <math.h>

// ---------------------------------------------------------------------------
// TranslationLoss (masked NLL == KLDiv-vs-one-hot, reduction='sum')
//   inp:    (4096, 32000) f32, target: (4096,) int
//   out:    scalar f32 = sum over rows t!=0 of (max + log(sum exp(x-max)) - x[t])
//
// MI455X plan:
//   * bandwidth-bound: 524 MB @ 23.3 TB/s => ~22.5 us floor; read HBM once.
//   * one workgroup (320 thr = 10 waves) per row; stage the 128 KB row into
//     LDS via CDNA5 async DMA (global_load_async_to_lds_b128 / ASYNCcnt).
//     IOFFSET is applied to BOTH the LDS and global address, so each chunk
//     needs zero address math: one base pair + offset:k*5120.
//   * software-pipelined: ASYNCcnt decrements in issue order, so chunk k is
//     consumable after s_wait_asynccnt <= 24-k; max pass overlaps the DMA.
//   * 25 chunks/thread exactly (8000 float4 / 320) -> uniform loops.
//     128 KB + 1.25 KB LDS -> 2 blocks per 320 KB WGP.
//   * deterministic 2-stage reduction (row losses -> d_ws -> tree sum).
// ---------------------------------------------------------------------------

#define V_DIM   32000
#define N_ROWS  4096
#define TPB     320                    // 10 waves (wave32)
#define VEC4    (V_DIM / 4)            // 8000 float4 chunks per row
#define CHUNKS  (VEC4 / TPB)           // 25 chunks per thread, exact
#define CHUNK_BYTES (TPB * 16)         // 5120 B stride between a thread's chunks
#define ROW_LDS_BYTES (V_DIM * 4)      // 128000 bytes dynamic LDS

__device__ __forceinline__ unsigned lds_offset_of(const void* p) {
    // Generic LDS pointers: SHARED_BASE in addr[63:32], LDS byte offset in
    // addr[31:0] -> truncation yields the workgroup-relative LDS address.
    return (unsigned)(unsigned long long)p;
}

// Issue chunk K: IOFFSET (compile-time) offsets both LDS dest and global src.
template <int K>
__device__ __forceinline__ void issue_async(unsigned lds_addr,
                                            unsigned long long gaddr) {
    if constexpr (K < CHUNKS) {
        asm volatile("global_load_async_to_lds_b128 %0, %1, off offset:%2"
                     :: "v"(lds_addr), "v"(gaddr), "i"(K * CHUNK_BYTES)
                     : "memory");
        issue_async<K + 1>(lds_addr, gaddr);
    }
}

// Pipelined consume: chunk K's async write has retired once the in-order
// ASYNCcnt drops to CHUNKS-1-K; then fold it into the running max.
template <int K>
__device__ __forceinline__ void staged_max(const float4* __restrict__ s4,
                                           int tid, float& m) {
    if constexpr (K < CHUNKS) {
        asm volatile("s_wait_asynccnt %0" :: "i"(CHUNKS - 1 - K) : "memory");
        float4 v = s4[tid + K * TPB];
        m = fmaxf(m, fmaxf(fmaxf(v.x, v.y), fmaxf(v.z, v.w)));
        staged_max<K + 1>(s4, tid, m);
    }
}

__global__ __launch_bounds__(TPB)
void row_nll_kernel(const float* __restrict__ inp,
                    const int*   __restrict__ target,
                    float*       __restrict__ row_loss) {
    extern __shared__ float s_row[];      // V_DIM floats (dynamic, 128 KB)
    __shared__ float s_red[TPB];

    const int row = blockIdx.x;
    const int tid = threadIdx.x;
    const float* g_row = inp + (size_t)row * V_DIM;

    // ---- Issue all async global->LDS copies (25 per thread, in order) ----
    const unsigned lds_addr = lds_offset_of(s_row) + (unsigned)tid * 16u;
    const unsigned long long gaddr =
        (unsigned long long)(const void*)(g_row + (size_t)tid * 4);
    issue_async<0>(lds_addr, gaddr);

    const float4* s_row4 = (const float4*)s_row;

    // ---- Pass 1: row max, overlapped with in-flight DMA ----
    float m = -INFINITY;
    staged_max<0>(s_row4, tid, m);        // ends with asynccnt == 0

    s_red[tid] = m;
    __syncthreads();                       // whole row now resident in LDS
    if (tid < 64) {
        float v = s_red[tid];
#pragma unroll
        for (int j = 1; j < 5; ++j) v = fmaxf(v, s_red[tid + 64 * j]);
        s_red[tid] = v;
    }
    __syncthreads();
    for (int off = 32; off > 0; off >>= 1) {
        if (tid < off) s_red[tid] = fmaxf(s_red[tid], s_red[tid + off]);
        __syncthreads();
    }
    m = s_red[0];
    __syncthreads();

    // ---- Pass 2: sum exp(x - max) from LDS (v_exp_f32) ----
    float s = 0.0f;
#pragma unroll
    for (int k = 0; k < CHUNKS; ++k) {
        float4 v = s_row4[tid + k * TPB];
        s += __expf(v.x - m) + __expf(v.y - m)
           + __expf(v.z - m) + __expf(v.w - m);
    }
    s_red[tid] = s;
    __syncthreads();
    if (tid < 64) {
        float v = s_red[tid];
#pragma unroll
        for (int j = 1; j < 5; ++j) v += s_red[tid + 64 * j];
        s_red[tid] = v;
    }
    __syncthreads();
    for (int off = 32; off > 0; off >>= 1) {
        if (tid < off) s_red[tid] += s_red[tid + off];
        __syncthreads();
    }

    if (tid == 0) {
        int t = target[row];
        float loss = 0.0f;
        if (t != 0) {                       // PADDING_IDX == 0 masked out
            loss = m + __logf(s_red[0]) - s_row[t];
        }
        row_loss[row] = loss;
    }
}

// Deterministic final reduction of the 4096 per-row losses.
__global__ __launch_bounds__(256)
void final_reduce_kernel(const float* __restrict__ row_loss,
                         float* __restrict__ out) {
    __shared__ float s_red[256];
    const int tid = threadIdx.x;
    float s = 0.0f;
    for (int i = tid; i < N_ROWS; i += 256) s += row_loss[i];  // fixed order
    s_red[tid] = s;
    __syncthreads();
    for (int off = 128; off > 0; off >>= 1) {
        if (tid < off) s_red[tid] += s_red[tid + off];
        __syncthreads();
    }
    if (tid == 0) out[0] = s_red[0];
}

extern "C" void kernel_launch(void* const* d_in, const int* in_sizes, int n_in,
                              void* d_out, int out_size, void* d_ws, size_t ws_size,
                              hipStream_t stream) {
    const float* inp    = (const float*)d_in[0];
    const int*   target = (const int*)  d_in[1];
    float*       out    = (float*)d_out;
    float*       row_ls = (float*)d_ws;          // 4096 floats of scratch

    row_nll_kernel<<<N_ROWS, TPB, ROW_LDS_BYTES, stream>>>(inp, target, row_ls);
    final_reduce_kernel<<<1, 256, 0, stream>>>(row_ls, out);
}